// pt_MultiScaleRoIAlign_41764261987335
// MI455X (gfx1250) — compile-verified
//
#include <hip/hip_runtime.h>
#include <math.h>

// ---------------------------------------------------------------------------
// Multiscale ROIAlign (FPN) for MI455X / gfx1250.
//
// Latency-bound gather op (0.2 GFLOP vs ~135MB HBM traffic; the 109MB feature
// pyramid is resident in the 192MB L2). WMMA is provably counterproductive:
// the separable interpolation matrices are 4-wide block-diagonal, giving <7%
// useful MACs per v_wmma_f32_16x16x4 -> slower than VALU FMA. The CDNA5
// features that DO apply: async global->LDS staging (ASYNCcnt path),
// global_prefetch_b8 patch warming, wave32/WGP-sized blocking.
// ---------------------------------------------------------------------------

#define ROI_OUT_H 7
#define ROI_OUT_W 7
#define NBIN      49          // 7*7 bins per ROI
#define NSAMP_X   14          // OUT_W * SR
#define NSAMP     196         // 49 bins * 4 samples (1:1, no sharing at SR=2)
#define NCH       256

__global__ __launch_bounds__(256)
void roi_align_fpn_kernel(const float* __restrict__ f0,
                          const float* __restrict__ f1,
                          const float* __restrict__ f2,
                          const float* __restrict__ f3,
                          const float* __restrict__ boxes,
                          float* __restrict__ out,
                          int rois_per_batch)
{
    __shared__ float  s_box[4];
    __shared__ int4   s_boff4[NBIN][4];  // [bin][sample] 4 tap offsets (y*W+x)
    __shared__ float4 s_bw4[NBIN][4];    // [bin][sample] 4 tap weights

    const int k   = blockIdx.x;          // ROI index
    const int tid = threadIdx.x;

    // ---- CDNA5 async global->LDS staging of this ROI's box (4 floats) -----
    if (tid < 4) {
        unsigned lds_off =
            (unsigned)(size_t)(&s_box[tid]);
        unsigned long long gaddr =
            (unsigned long long)(size_t)(boxes + (size_t)k * 4 + tid);
        asm volatile("global_load_async_to_lds_b32 %0, %1, off"
                     :: "v"(lds_off), "v"(gaddr)
                     : "memory");
    }
    asm volatile("s_wait_asynccnt 0" ::: "memory");
    __syncthreads();

    const float x1b = s_box[0];
    const float y1b = s_box[1];
    const float x2b = s_box[2];
    const float y2b = s_box[3];

    // ---- FPN level selection (uniform across the workgroup) ---------------
    const float area = (x2b - x1b) * (y2b - y1b);
    const float ssz  = sqrtf(area);
    float lvlf = floorf(4.0f + log2f(ssz / 224.0f) + 1e-6f);
    lvlf = fminf(fmaxf(lvlf, 2.0f), 5.0f);
    const int li = (int)lvlf - 2;        // 0..3

    const float* fp; int H, W; float sc;
    switch (li) {
        case 0:  fp = f0; H = 200; W = 200; sc = 0.25f;    break;
        case 1:  fp = f1; H = 100; W = 100; sc = 0.125f;   break;
        case 2:  fp = f2; H =  50; W =  50; sc = 0.0625f;  break;
        default: fp = f3; H =  25; W =  25; sc = 0.03125f; break;
    }

    const int    b       = k / rois_per_batch;  // batch index
    const size_t planeHW = (size_t)H * (size_t)W;
    const float* fb      = fp + (size_t)b * NCH * planeHW;

    // ---- ROI geometry (matches reference arithmetic exactly) --------------
    const float x1s = x1b * sc, y1s = y1b * sc;
    const float x2s = x2b * sc, y2s = y2b * sc;
    const float roi_w = fmaxf(x2s - x1s, 1.0f);
    const float roi_h = fmaxf(y2s - y1s, 1.0f);
    const float bin_w = roi_w / 7.0f;
    const float bin_h = roi_h / 7.0f;

    // Warm this channel's patch start in L2/L0 while tap tables are built.
    {
        int py0 = (int)fmaxf(y1s, 0.0f); if (py0 > H - 1) py0 = H - 1;
        int px0 = (int)fmaxf(x1s, 0.0f); if (px0 > W - 1) px0 = W - 1;
        const float* pfetch = fb + (size_t)tid * planeHW + (size_t)py0 * W + px0;
        __builtin_prefetch(pfetch, 0, 1);   // -> global_prefetch_b8
    }

    // ---- Build per-(bin,sample) tap tables --------------------------------
    // Threads 0..195: u = bin*4 + j, j = (iy,ix) in {0,1}^2.
    if (tid < NSAMP) {
        const int bin = tid >> 2;
        const int j   = tid & 3;
        const int ph  = bin / ROI_OUT_W;
        const int pw  = bin - ph * ROI_OUT_W;
        const int iy  = j >> 1;
        const int ix  = j & 1;

        float y = y1s + (float)ph * bin_h + ((float)iy + 0.5f) * (bin_h / 2.0f);
        float x = x1s + (float)pw * bin_w + ((float)ix + 0.5f) * (bin_w / 2.0f);

        // _prep_coords (y)
        y = fmaxf(y, 0.0f);
        int ylo = (int)y;
        const bool yedge = (ylo >= H - 1);
        const int yhi = yedge ? (H - 1) : (ylo + 1);
        ylo = yedge ? (H - 1) : ylo;
        const float ly = y - (float)ylo;
        const float hy = 1.0f - ly;

        // _prep_coords (x)
        x = fmaxf(x, 0.0f);
        int xlo = (int)x;
        const bool xedge = (xlo >= W - 1);
        const int xhi = xedge ? (W - 1) : (xlo + 1);
        xlo = xedge ? (W - 1) : xlo;
        const float lx = x - (float)xlo;
        const float hx = 1.0f - lx;

        int4 off;
        off.x = ylo * W + xlo;
        off.y = ylo * W + xhi;
        off.z = yhi * W + xlo;
        off.w = yhi * W + xhi;
        s_boff4[bin][j] = off;

        float4 w;
        w.x = hy * hx;
        w.y = hy * lx;
        w.z = ly * hx;
        w.w = ly * lx;
        s_bw4[bin][j] = w;
    }
    __syncthreads();

    // ---- Main gather loop -------------------------------------------------
    // 49 outputs per thread; unrolled x2 -> 32 independent scattered loads in
    // flight per wave per step (latency-bound L2 hits). Stores are coalesced.
    float* ob = out + (size_t)k * (NCH * NBIN);

    const int total = NCH * NBIN;                 // 12544
    int o = tid;
    for (; o + 256 < total; o += 512) {
        const int o2   = o + 256;
        const int c1   = o  / NBIN;
        const int c2   = o2 / NBIN;
        const int bin1 = o  - c1 * NBIN;
        const int bin2 = o2 - c2 * NBIN;
        const float* p1 = fb + (size_t)c1 * planeHW;
        const float* p2 = fb + (size_t)c2 * planeHW;

        float acc1 = 0.0f, acc2 = 0.0f;
#pragma unroll
        for (int j = 0; j < 4; ++j) {
            const int4   off1 = s_boff4[bin1][j];   // ds_load_b128
            const float4 w1   = s_bw4[bin1][j];     // ds_load_b128
            const int4   off2 = s_boff4[bin2][j];
            const float4 w2   = s_bw4[bin2][j];
            acc1 += w1.x * p1[off1.x];
            acc1 += w1.y * p1[off1.y];
            acc1 += w1.z * p1[off1.z];
            acc1 += w1.w * p1[off1.w];
            acc2 += w2.x * p2[off2.x];
            acc2 += w2.y * p2[off2.y];
            acc2 += w2.z * p2[off2.z];
            acc2 += w2.w * p2[off2.w];
        }
        ob[o]  = acc1 * 0.25f;   // / (SR*SR)
        ob[o2] = acc2 * 0.25f;
    }
    // Tail (49 iterations is odd -> one remaining output per thread)
    for (; o < total; o += 256) {
        const int c   = o / NBIN;
        const int bin = o - c * NBIN;
        const float* pc = fb + (size_t)c * planeHW;

        float acc = 0.0f;
#pragma unroll
        for (int j = 0; j < 4; ++j) {
            const int4   off = s_boff4[bin][j];
            const float4 w   = s_bw4[bin][j];
            acc += w.x * pc[off.x];
            acc += w.y * pc[off.y];
            acc += w.z * pc[off.z];
            acc += w.w * pc[off.w];
        }
        ob[o] = acc * 0.25f;
    }
}

extern "C" void kernel_launch(void* const* d_in, const int* in_sizes, int n_in,
                              void* d_out, int out_size, void* d_ws, size_t ws_size,
                              hipStream_t stream) {
    const float* f0    = (const float*)d_in[0];   // (B,256,200,200)
    const float* f1    = (const float*)d_in[1];   // (B,256,100,100)
    const float* f2    = (const float*)d_in[2];   // (B,256, 50, 50)
    const float* f3    = (const float*)d_in[3];   // (B,256, 25, 25)
    const float* boxes = (const float*)d_in[4];   // (B,L,4)

    const int K = in_sizes[4] / 4;                         // total ROIs = B*L
    int B = in_sizes[0] / (NCH * 200 * 200);
    if (B <= 0) B = 1;
    const int L = K / B;                                   // ROIs per batch

    roi_align_fpn_kernel<<<dim3(K), dim3(256), 0, stream>>>(
        f0, f1, f2, f3, boxes, (float*)d_out, L);
}